// GraphConvolutionSparse_60335700574617
// MI455X (gfx1250) — compile-verified
//
#include <hip/hip_runtime.h>

typedef __attribute__((ext_vector_type(2))) float v2f;
typedef __attribute__((ext_vector_type(8))) float v8f;

#define N_NODES 100000
#define N_EDGES 3200000
#define IN_DIM  256
#define UNITS   128

// ---------------------------------------------------------------------------
// Phase 1: h = x @ W  via V_WMMA_F32_16X16X4_F32 (fp32-native matrix pipe).
// One wave computes one 16x16 tile of h. Block = 256 threads = 8 waves:
//   block b   -> row tile  [16b, 16b+16)
//   wave  w   -> col tile  [16w, 16w+16)   (UNITS/16 == 8 waves exactly)
// A (16x4 f32) per-lane layout (ISA 7.12.2): lanes 0-15 hold K=0(v0),K=1(v1);
// lanes 16-31 hold K=2(v0),K=3(v1); M = lane%16.
// B (4x16) striped symmetrically: lane%16 = N column, same K split.
// C/D (16x16 f32): VGPR j -> M = j + 8*(lane/16), N = lane%16.
// ---------------------------------------------------------------------------
__global__ __launch_bounds__(256) void gcn_gemm_wmma_f32(
    const float* __restrict__ x, const float* __restrict__ w,
    float* __restrict__ h)
{
    const int wave = threadIdx.x >> 5;   // 0..7  -> n tile
    const int lane = threadIdx.x & 31;   // 0..31
    const int half = lane >> 4;          // 0/1  -> K sub-pair
    const int m    = lane & 15;          // row within tile / col within tile
    const int row0 = blockIdx.x * 16;
    const int col0 = wave * 16;

    const float* xrow = x + (size_t)(row0 + m) * IN_DIM;
    const float* wcol = w + (size_t)(col0 + m);

    v8f c = {};
    #pragma unroll 8
    for (int k = 0; k < IN_DIM; k += 4) {
        const int ka = k + 2 * half;
        v2f a, b;
        a.x = xrow[ka];                       // contiguous pair -> b64 load
        a.y = xrow[ka + 1];
        b.x = wcol[(size_t)ka * UNITS];       // W[ka][col0+m]
        b.y = wcol[(size_t)(ka + 1) * UNITS];
        // 8 args: (neg_a, A, neg_b, B, c_mod, C, reuse_a, reuse_b)
        c = __builtin_amdgcn_wmma_f32_16x16x4_f32(
            false, a, false, b, (short)0, c, false, false);
    }

    float* hout = h + (size_t)row0 * UNITS + col0 + m;
    #pragma unroll
    for (int j = 0; j < 8; ++j) {
        hout[(size_t)(j + 8 * half) * UNITS] = c[j];
    }
}

// ---------------------------------------------------------------------------
// Phase 2a: zero the accumulator (harness poisons d_out with 0xAA).
// ---------------------------------------------------------------------------
__global__ __launch_bounds__(256) void gcn_zero(float4* __restrict__ out, int n4)
{
    const int i = blockIdx.x * blockDim.x + threadIdx.x;
    if (i < n4) out[i] = make_float4(0.f, 0.f, 0.f, 0.f);
}

// ---------------------------------------------------------------------------
// Phase 2b: edge scatter. One wave per edge: 32 lanes x float4 = one fully
// coalesced 512B gather of h[col], then 4 non-returning global_atomic_add_f32
// per lane into out[row] (inline asm guarantees the native L2 atomic; no CAS
// fallback). h (51 MB) and out (51 MB) both live in the 192 MB L2, so the
// gathers and atomics resolve at L2, not HBM. The edge index e is wave-
// uniform: readfirstlane makes it an SGPR so rows/cols/vals become scalar
// (constant-cache) loads instead of 32-lane redundant vector loads.
// Grid is exact (N_EDGES/8 blocks of 8 waves) -> no bounds branch, EXEC
// stays all-ones.
// ---------------------------------------------------------------------------
__device__ __forceinline__ void atomic_add_f32_l2(float* p, float v)
{
    asm volatile("global_atomic_add_f32 %0, %1, off"
                 :
                 : "v"(p), "v"(v)
                 : "memory");
}

__global__ __launch_bounds__(256) void gcn_spmm_scatter(
    const float* __restrict__ h,
    const int*   __restrict__ rows,
    const int*   __restrict__ cols,
    const float* __restrict__ vals,
    float*       __restrict__ out)
{
    // 8 waves per block, one edge per wave; uniform -> SGPR.
    const int e = __builtin_amdgcn_readfirstlane(
        blockIdx.x * 8 + (threadIdx.x >> 5));
    const int u = (threadIdx.x & 31) * 4;     // unit chunk 0,4,...,124

    const int   r = rows[e];                  // scalar loads (uniform addr)
    const int   s = cols[e];
    const float v = vals[e];

    const float4 hv = *(const float4*)(h + (size_t)s * UNITS + u);
    float* o = out + (size_t)r * UNITS + u;
    atomic_add_f32_l2(o + 0, v * hv.x);
    atomic_add_f32_l2(o + 1, v * hv.y);
    atomic_add_f32_l2(o + 2, v * hv.z);
    atomic_add_f32_l2(o + 3, v * hv.w);
}

// ---------------------------------------------------------------------------
// Phase 3: ReLU in-place on out.
// ---------------------------------------------------------------------------
__global__ __launch_bounds__(256) void gcn_relu(float4* __restrict__ out, int n4)
{
    const int i = blockIdx.x * blockDim.x + threadIdx.x;
    if (i < n4) {
        float4 x = out[i];
        x.x = fmaxf(x.x, 0.f);
        x.y = fmaxf(x.y, 0.f);
        x.z = fmaxf(x.z, 0.f);
        x.w = fmaxf(x.w, 0.f);
        out[i] = x;
    }
}

extern "C" void kernel_launch(void* const* d_in, const int* in_sizes, int n_in,
                              void* d_out, int out_size, void* d_ws, size_t ws_size,
                              hipStream_t stream)
{
    const float* x     = (const float*)d_in[0];   // [N_NODES, IN_DIM]
    const float* w     = (const float*)d_in[1];   // [IN_DIM, UNITS]
    const int*   rows  = (const int*)  d_in[2];   // [N_EDGES]
    const int*   cols  = (const int*)  d_in[3];   // [N_EDGES]
    const float* vals  = (const float*)d_in[4];   // [N_EDGES]
    float*       out   = (float*)d_out;           // [N_NODES, UNITS]
    float*       h     = (float*)d_ws;            // [N_NODES, UNITS] scratch

    // Phase 1: dense transform h = x @ W (fp32 WMMA).
    // 100000 rows / 16 = 6250 blocks; 8 waves = 8 column tiles of 16.
    gcn_gemm_wmma_f32<<<N_NODES / 16, 256, 0, stream>>>(x, w, h);

    // Phase 2a: zero accumulator.
    const int n4 = (N_NODES * UNITS) / 4;   // 3.2M float4
    gcn_zero<<<(n4 + 255) / 256, 256, 0, stream>>>((float4*)out, n4);

    // Phase 2b: scatter-add over edges (one wave per edge, 8 edges/block).
    gcn_spmm_scatter<<<N_EDGES / 8, 256, 0, stream>>>(h, rows, cols, vals, out);

    // Phase 3: ReLU.
    gcn_relu<<<(n4 + 255) / 256, 256, 0, stream>>>((float4*)out, n4);
}